// EvalMultiModel_43190191128870
// MI455X (gfx1250) — compile-verified
//
#include <hip/hip_runtime.h>

// Problem constants (match reference)
#define MM  2
#define DD  128
#define NIc 131072
#define LLc 4
#define NPLc 2048
#define RRc 8
#define TTc 32
#define NNc (NIc + LLc * NPLc)   // 139264 = 544*256 = 2176*64

typedef __attribute__((ext_vector_type(2))) float v2f;
typedef __attribute__((ext_vector_type(8))) float v8f;
typedef __attribute__((ext_vector_type(4))) unsigned int u32x4;
typedef __attribute__((ext_vector_type(8))) int i32x8;
typedef __attribute__((ext_vector_type(4))) int i32x4;

__device__ __forceinline__ v8f wmma_f32(v2f a, v2f b, v8f c) {
  // D = A(16x4) * B(4x16) + C(16x16), fp32
  return __builtin_amdgcn_wmma_f32_16x16x4_f32(false, a, false, b, (short)0, c,
                                               false, false);
}

// ---------------------------------------------------------------------------
// Kernel 1: init embedding  store[m, i, :] = init_table[m, thax[i], :] + sine[i]*s_vec[m, :]
__global__ void k_init_emb(const int* __restrict__ thax, const float* __restrict__ sine,
                           const float* __restrict__ init_table, const float* __restrict__ s_vec,
                           float* __restrict__ store) {
  int t = threadIdx.x;
  int lane = t & 31;
  int rowInBlk = t >> 5;
  long long gid = (long long)blockIdx.x * 8 + rowInBlk;   // over M*NI
  int m = (int)(gid / NIc);
  int i = (int)(gid % NIc);
  int tx = thax[i];
  float sv = sine[i];
  const float4* it  = (const float4*)(init_table + (long long)(m * TTc + tx) * DD);
  const float4* svv = (const float4*)(s_vec + m * DD);
  float4 a = it[lane];
  float4 b = svv[lane];
  float4 o;
  o.x = a.x + sv * b.x; o.y = a.y + sv * b.y;
  o.z = a.z + sv * b.z; o.w = a.w + sv * b.w;
  ((float4*)(store + ((long long)m * NNc + i) * DD))[lane] = o;
}

// ---------------------------------------------------------------------------
// Counting sort of nodes by rule, per layer (shared across models)
__global__ void k_zero_counts(int* __restrict__ cnt) {
  if (threadIdx.x < LLc * RRc) cnt[threadIdx.x] = 0;
}
__global__ void k_bucket(const int* __restrict__ ridx, int* __restrict__ cnt,
                         int* __restrict__ lists) {
  int j = blockIdx.x * blockDim.x + threadIdx.x;   // 0 .. L*NPL-1
  int l = j / NPLc, jj = j % NPLc;
  int r = ridx[j];
  int slot = atomicAdd(&cnt[l * RRc + r], 1);
  lists[(l * RRc + r) * NPLc + slot] = jj;
}

// ---------------------------------------------------------------------------
// Kernel 3: one DAG layer. grid = (NPL/32 tiles, R, M), block = 256 (8 waves).
// 32 nodes of rule r, model m per block (2 row-tiles per wave, B reused).
__global__ void k_deriv(int l, const int* __restrict__ par, const int* __restrict__ cnt,
                        const int* __restrict__ lists,
                        const float* __restrict__ W1, const float* __restrict__ b1,
                        const float* __restrict__ W2, const float* __restrict__ b2,
                        const float* __restrict__ tweaks, float* __restrict__ store) {
  __shared__ float catb[32 * 256];   // 32 KB
  __shared__ float hbuf[32 * 128];   // 16 KB
  __shared__ int sj[32], sp0[32], sp1[32];

  int r = blockIdx.y;
  int m = blockIdx.z;
  int c = cnt[l * RRc + r];
  int start = blockIdx.x * 32;
  if (start >= c) return;   // uniform across the block

  int t = threadIdx.x;
  if (t < 32) {
    int idx = start + t;
    if (idx >= c) idx = c - 1;     // clamp; padded rows computed, not scattered
    int jj = lists[(l * RRc + r) * NPLc + idx];
    sj[t]  = jj;
    sp0[t] = par[(l * NPLc + jj) * 2 + 0];
    sp1[t] = par[(l * NPLc + jj) * 2 + 1];
  }
  __syncthreads();

  // Gather cat tile: 8 threads/row, 32 floats each (8x float4)
  {
    int row = t >> 3, sub = t & 7;
    const float* src = (sub < 4)
        ? (store + ((long long)m * NNc + sp0[row]) * DD + sub * 32)
        : (store + ((long long)m * NNc + sp1[row]) * DD + (sub - 4) * 32);
    const float4* s4 = (const float4*)src;
    float4* dst = (float4*)(catb + row * 256 + sub * 32);
#pragma unroll
    for (int i = 0; i < 8; ++i) dst[i] = s4[i];
  }
  __syncthreads();

  int wave = t >> 5, lane = t & 31;
  int nrow = lane & 15, hi = lane >> 4;
  int col = wave * 16 + nrow;

  const float* W1p = W1 + ((long long)(m * RRc + r) * 256) * 128;
  const float* W2p = W2 + ((long long)(m * RRc + r) * 128) * 128;
  __builtin_prefetch(W2p + col, 0, 3);   // global_prefetch_b8 for 2nd GEMM

  // GEMM 1: [32 x 256] x [256 x 128]; each B fragment reused by 2 row-tiles
  v8f acc[2] = {{}, {}};
  for (int kk = 0; kk < 64; ++kk) {
    int k0 = kk * 4 + 2 * hi;
    v2f b;
    b.x = W1p[(long long)k0 * 128 + col];               // B: K=k0,   N=col
    b.y = W1p[(long long)(k0 + 1) * 128 + col];         // B: K=k0+1, N=col
#pragma unroll
    for (int rt = 0; rt < 2; ++rt) {
      v2f a = *(const v2f*)(catb + (rt * 16 + nrow) * 256 + k0);
      acc[rt] = wmma_f32(a, b, acc[rt]);
    }
  }
  float bb = b1[(long long)(m * RRc + r) * 128 + col];
#pragma unroll
  for (int rt = 0; rt < 2; ++rt) {
#pragma unroll
    for (int v = 0; v < 8; ++v) {
      float x = acc[rt][v] + bb;
      x = x > 0.f ? x : 0.f;
      hbuf[(rt * 16 + v + 8 * hi) * 128 + col] = x;
    }
  }
  __syncthreads();

  // GEMM 2: [32 x 128] x [128 x 128]
  v8f acc2[2] = {{}, {}};
  for (int kk = 0; kk < 32; ++kk) {
    int k0 = kk * 4 + 2 * hi;
    v2f b;
    b.x = W2p[(long long)k0 * 128 + col];
    b.y = W2p[(long long)(k0 + 1) * 128 + col];
#pragma unroll
    for (int rt = 0; rt < 2; ++rt) {
      v2f a = *(const v2f*)(hbuf + (rt * 16 + nrow) * 128 + k0);
      acc2[rt] = wmma_f32(a, b, acc2[rt]);
    }
  }
  float b2v = b2[(long long)(m * RRc + r) * 128 + col] + tweaks[m * 2 + 1];
  int gbase = NIc + l * NPLc;
#pragma unroll
  for (int rt = 0; rt < 2; ++rt) {
#pragma unroll
    for (int v = 0; v < 8; ++v) {
      int row = rt * 16 + v + 8 * hi;
      if (start + row < c) {
        int g = gbase + sj[row];
        store[((long long)m * NNc + g) * DD + col] = acc2[rt][v] + b2v;
      }
    }
  }
}

// ---------------------------------------------------------------------------
// Kernel 4: eval_net. grid = (N/64, M), block 256 (8 waves).
// A tile (64 nodes x 128) staged via the Tensor Data Mover when available:
// one tensor_load_to_lds moves the whole 32 KB tile, tracked by TENSORcnt.
__global__ void k_eval(const float* __restrict__ store, const float* __restrict__ We1,
                       const float* __restrict__ be1, const float* __restrict__ we2,
                       const float* __restrict__ be2, const float* __restrict__ tweaks,
                       float* __restrict__ logits) {
  __shared__ float ab[64 * 128];   // 32 KB: A tile, later reused for he*we2 products
  int m = blockIdx.y;
  long long g0 = (long long)blockIdx.x * 64;
  int t = threadIdx.x;

#if defined(__gfx1250__) && __has_builtin(__builtin_amdgcn_tensor_load_to_lds)
  if (t < 32) {   // one wave issues the TDM op (EXEC ignored by tensor ops)
    unsigned lds_addr =
        (unsigned)(unsigned long long)(__attribute__((address_space(3))) void*)ab;
    unsigned long long ga =
        (unsigned long long)(const float*)(store + ((long long)m * NNc + g0) * DD);
    // D# group 0: count=1, lds_addr, global_addr[56:0], type=2
    u32x4 g0d;
    g0d.x = 1u;                                        // count=1, no gather
    g0d.y = lds_addr;                                  // LDS byte address
    g0d.z = (unsigned)(ga & 0xFFFFFFFFu);              // global_addr[31:0]
    g0d.w = (unsigned)((ga >> 32) & 0x01FFFFFFu) | 0x80000000u;  // [56:32] | type=2
    // D# group 1: data_size=4B, tensor 128x64, tile 128x64, stride0=128
    i32x8 g1d;
    g1d[0] = 2 << 16;            // data_size = 2 (4 bytes); wg_mask=0
    g1d[1] = 128 << 16;          // tensor_dim0 low16 = 128 (bits 63:48)
    g1d[2] = 64 << 16;           // tensor_dim0 hi16=0 | tensor_dim1 low16 = 64
    g1d[3] = 128 << 16;          // tensor_dim1 hi16=0 | tile_dim0 = 128
    g1d[4] = 64;                 // tile_dim1 = 64 | tile_dim2 = 0
    g1d[5] = 128;                // tensor_dim0_stride low32 = 128
    g1d[6] = 0;                  // stride0 hi16 | tensor_dim1_stride low16
    g1d[7] = 0;                  // tensor_dim1_stride hi
    i32x4 g2d = {0, 0, 0, 0};    // unused (2D tile)
    i32x4 g3d = {0, 0, 0, 0};
    i32x8 g4d = {0, 0, 0, 0, 0, 0, 0, 0};   // extra group (clang-23 6-arg form)
    __builtin_amdgcn_tensor_load_to_lds(g0d, g1d, g2d, g3d, g4d, 0);
    __builtin_amdgcn_s_wait_tensorcnt(0);   // drain TENSORcnt in issuing wave
  }
  __syncthreads();
#else
  {
    int row = t >> 2, sub = t & 3;   // 4 threads/row, 32 floats each
    const float4* s4 =
        (const float4*)(store + ((long long)m * NNc + g0 + row) * DD + sub * 32);
    float4* dst = (float4*)(ab + row * 128 + sub * 32);
#pragma unroll
    for (int i = 0; i < 8; ++i) dst[i] = s4[i];
  }
  __syncthreads();
#endif

  int wave = t >> 5, lane = t & 31;
  int nrow = lane & 15, hi = lane >> 4;
  int col = wave * 16 + nrow;
  const float* Wp = We1 + (long long)m * 128 * 128;

  v8f acc[4] = {{}, {}, {}, {}};
  for (int kk = 0; kk < 32; ++kk) {
    int k0 = kk * 4 + 2 * hi;
    v2f b;
    b.x = Wp[(long long)k0 * 128 + col];
    b.y = Wp[(long long)(k0 + 1) * 128 + col];
#pragma unroll
    for (int rt = 0; rt < 4; ++rt) {
      v2f a = *(const v2f*)(ab + (rt * 16 + nrow) * 128 + k0);
      acc[rt] = wmma_f32(a, b, acc[rt]);
    }
  }
  float bb  = be1[m * 128 + col];
  float w2v = we2[m * 128 + col];
  __syncthreads();   // everyone done reading ab
#pragma unroll
  for (int rt = 0; rt < 4; ++rt) {
#pragma unroll
    for (int v = 0; v < 8; ++v) {
      float x = acc[rt][v] + bb;
      x = x > 0.f ? x : 0.f;
      ab[(rt * 16 + v + 8 * hi) * 128 + col] = x * w2v;   // he*we2 products
    }
  }
  __syncthreads();
  if (t < 64) {   // fixed-order row reduction -> deterministic
    float s = 0.f;
    for (int k = 0; k < 128; ++k) s += ab[t * 128 + k];
    logits[(long long)m * NNc + g0 + t] = s + be2[m] + tweaks[m * 2 + 0];
  }
}

// ---------------------------------------------------------------------------
// Reductions: fixed-structure two-stage trees (deterministic, no fp atomics)
__global__ void k_reduce_totals(const float* __restrict__ pos, const float* __restrict__ neg,
                                float* __restrict__ pp, float* __restrict__ pn) {
  __shared__ float sp[256], sn[256];
  int i = blockIdx.x * 256 + threadIdx.x;
  sp[threadIdx.x] = pos[i];
  sn[threadIdx.x] = neg[i];
  __syncthreads();
  for (int s = 128; s > 0; s >>= 1) {
    if (threadIdx.x < s) {
      sp[threadIdx.x] += sp[threadIdx.x + s];
      sn[threadIdx.x] += sn[threadIdx.x + s];
    }
    __syncthreads();
  }
  if (threadIdx.x == 0) { pp[blockIdx.x] = sp[0]; pn[blockIdx.x] = sn[0]; }
}

__global__ void k_final2(const float* __restrict__ pp, const float* __restrict__ pn,
                         float* __restrict__ out) {
  __shared__ float sp[256], sn[256];
  int t = threadIdx.x;
  float a = 0.f, b = 0.f;
  for (int i = t; i < NNc / 256; i += 256) { a += pp[i]; b += pn[i]; }
  sp[t] = a; sn[t] = b;
  __syncthreads();
  for (int s = 128; s > 0; s >>= 1) {
    if (t < s) { sp[t] += sp[t + s]; sn[t] += sn[t + s]; }
    __syncthreads();
  }
  if (t == 0) { out[6] = sp[0]; out[7] = sn[0]; }
}

__device__ __forceinline__ float log_sigmoid_f(float x) {
  return fminf(x, 0.f) - log1pf(expf(-fabsf(x)));
}

__global__ void k_bce(const float* __restrict__ logits, const float* __restrict__ pos,
                      const float* __restrict__ neg, const float* __restrict__ totals,
                      float* __restrict__ part) {
  __shared__ float sh[256];
  int i = blockIdx.x * 256 + threadIdx.x;
  float p = pos[i], q = neg[i];
  float pw = totals[7] / totals[6];    // tot_neg / tot_pos
  float gold = p / (p + q);
  float w = p + q;
  float vals[6];
#pragma unroll
  for (int m = 0; m < 2; ++m) {
    float lg = logits[(long long)m * NNc + i];
    float bce = -(pw * gold * log_sigmoid_f(lg) + (1.f - gold) * log_sigmoid_f(-lg));
    vals[m]     = w * bce;
    vals[2 + m] = (lg >= 0.f) ? p : 0.f;
    vals[4 + m] = (lg <  0.f) ? q : 0.f;
  }
  for (int qd = 0; qd < 6; ++qd) {
    sh[threadIdx.x] = vals[qd];
    __syncthreads();
    for (int s = 128; s > 0; s >>= 1) {
      if (threadIdx.x < s) sh[threadIdx.x] += sh[threadIdx.x + s];
      __syncthreads();
    }
    if (threadIdx.x == 0) part[qd * (NNc / 256) + blockIdx.x] = sh[0];
    __syncthreads();
  }
}

__global__ void k_final6(const float* __restrict__ part, float* __restrict__ out) {
  __shared__ float sh[256];
  int t = threadIdx.x;
  for (int qd = 0; qd < 6; ++qd) {
    float a = 0.f;
    for (int i = t; i < NNc / 256; i += 256) a += part[qd * (NNc / 256) + i];
    sh[t] = a;
    __syncthreads();
    for (int s = 128; s > 0; s >>= 1) {
      if (t < s) sh[t] += sh[t + s];
      __syncthreads();
    }
    if (t == 0) out[qd] = sh[0];
    __syncthreads();
  }
}

// ---------------------------------------------------------------------------
extern "C" void kernel_launch(void* const* d_in, const int* in_sizes, int n_in,
                              void* d_out, int out_size, void* d_ws, size_t ws_size,
                              hipStream_t stream) {
  (void)in_sizes; (void)n_in; (void)out_size; (void)ws_size;
  const int*   thax = (const int*)d_in[0];
  const float* sine = (const float*)d_in[1];
  const int*   par  = (const int*)d_in[2];
  const int*   ridx = (const int*)d_in[3];
  const float* pos  = (const float*)d_in[4];
  const float* neg  = (const float*)d_in[5];
  const float* init_table = (const float*)d_in[6];
  const float* s_vec = (const float*)d_in[7];
  const float* W1  = (const float*)d_in[8];
  const float* b1  = (const float*)d_in[9];
  const float* W2  = (const float*)d_in[10];
  const float* b2  = (const float*)d_in[11];
  const float* tweaks = (const float*)d_in[12];
  const float* We1 = (const float*)d_in[13];
  const float* be1 = (const float*)d_in[14];
  const float* we2 = (const float*)d_in[15];
  const float* be2 = (const float*)d_in[16];
  float* out = (float*)d_out;

  // Workspace layout (~144 MB)
  char* ws = (char*)d_ws;
  size_t off = 0;
  float* store  = (float*)(ws + off); off += (size_t)MM * NNc * DD * sizeof(float);
  float* logits = (float*)(ws + off); off += (size_t)MM * NNc * sizeof(float);
  int*   cnt    = (int*)(ws + off);   off += 64 * sizeof(int);
  int*   lists  = (int*)(ws + off);   off += (size_t)LLc * RRc * NPLc * sizeof(int);
  float* pp     = (float*)(ws + off); off += (NNc / 256) * sizeof(float);
  float* pn     = (float*)(ws + off); off += (NNc / 256) * sizeof(float);
  float* part6  = (float*)(ws + off); off += 6 * (NNc / 256) * sizeof(float);

  k_zero_counts<<<1, 64, 0, stream>>>(cnt);
  k_bucket<<<(LLc * NPLc) / 256, 256, 0, stream>>>(ridx, cnt, lists);
  k_init_emb<<<(MM * NIc) / 8, 256, 0, stream>>>(thax, sine, init_table, s_vec, store);
  for (int l = 0; l < LLc; ++l) {
    k_deriv<<<dim3(NPLc / 32, RRc, MM), 256, 0, stream>>>(
        l, par, cnt, lists, W1, b1, W2, b2, tweaks, store);
  }
  k_eval<<<dim3(NNc / 64, MM), 256, 0, stream>>>(store, We1, be1, we2, be2, tweaks, logits);
  k_reduce_totals<<<NNc / 256, 256, 0, stream>>>(pos, neg, pp, pn);
  k_final2<<<1, 256, 0, stream>>>(pp, pn, out);
  k_bce<<<NNc / 256, 256, 0, stream>>>(logits, pos, neg, out, part6);
  k_final6<<<1, 256, 0, stream>>>(part6, out);
}